// MetaGRU_83562883711141
// MI455X (gfx1250) — compile-verified
//
#include <hip/hip_runtime.h>
#include <cmath>

typedef __attribute__((ext_vector_type(2))) float v2f;
typedef __attribute__((ext_vector_type(8))) float v8f;

#define NN 50000
#define EE 800000
#define GG 64
#define HH 64
#define NSTEPS 3

// ---------------------------------------------------------------------------
// WMMA helper: for a 16-row tile, acc[t] += concat_s(rows_s) @ W_s^T
// A operand: per-lane row pointer (lane l supplies row M = l&15; lanes 16-31
//   duplicate the pointer and read the K=+2 pair per the 16x4 f32 A layout).
// B operand: weight rows (out = in @ W.T -> output col n uses W row n),
//   per-source base pointer + row stride.
// C/D: v8f accumulators per 16-col N tile (C/D layout: vgpr j -> M=j(+8),
//   lane&15 -> N).
// ---------------------------------------------------------------------------
struct MMSrc {
  const float* a;   // per-lane row pointer (64 K-values for this source)
  const float* w;   // weight base (row-major), rows indexed by output column
  int ldw;          // weight row stride in floats
};

template <int NSRC, int NTILES>
__device__ __forceinline__ void mm16(const MMSrc* s, int wrow0, v8f* acc,
                                     int sub, int koff) {
#pragma unroll
  for (int si = 0; si < NSRC; ++si) {
    const float* ap = s[si].a;
    const float* bp[NTILES];
#pragma unroll
    for (int t = 0; t < NTILES; ++t)
      bp[t] = s[si].w + (size_t)(wrow0 + t * 16 + sub) * s[si].ldw;
#pragma unroll
    for (int kk = 0; kk < 64; kk += 4) {
      v2f a = *(const v2f*)(ap + kk + koff);
#pragma unroll
      for (int t = 0; t < NTILES; ++t) {
        v2f b = *(const v2f*)(bp[t] + kk + koff);
        acc[t] = __builtin_amdgcn_wmma_f32_16x16x4_f32(
            false, a, false, b, (short)0, acc[t], false, false);
      }
    }
  }
}

__device__ __forceinline__ float sigmf(float x) {
  return 1.0f / (1.0f + __expf(-x));
}

// ---------------------------------------------------------------------------
// Edge step: e_out = relu([x[src],x[dst],ea,u[b[src]]] @ We.T + be)
//            ea'   = GRU(e_out, ea); atomic-accumulate ea' into agg[dst]
// One wave handles 16 consecutive edges.  x/u rows have stride 256 (history
// slabs in d_out); ea rows have stride 64.
// ---------------------------------------------------------------------------
__global__ __launch_bounds__(128) void edge_step_kernel(
    const float* __restrict__ x_cur, const float* __restrict__ u_cur,
    const float* __restrict__ ea_in, float* __restrict__ ea_out,
    float* __restrict__ agg, const int* __restrict__ src,
    const int* __restrict__ dst, const int* __restrict__ batch,
    const float* __restrict__ We, const float* __restrict__ be,
    const float* __restrict__ Wih, const float* __restrict__ Whh,
    const float* __restrict__ bih, const float* __restrict__ bhh) {
  __shared__ float lds[4][16][68];
  const int lane = threadIdx.x & 31;
  const int wv = threadIdx.x >> 5;
  const int sub = lane & 15;
  const int half = lane >> 4;
  const int koff = half * 2;
  const int tile = blockIdx.x * 4 + wv;
  if (tile * 16 >= EE) return;
  const int e0 = tile * 16;
  const int es = e0 + sub;
  const int si = src[es];
  const int di = dst[es];
  const int gi = batch[si];

  // ---- edge linear + relu ----
  v8f acc[4];
#pragma unroll
  for (int t = 0; t < 4; ++t) {
    float bv = be[t * 16 + sub];
#pragma unroll
    for (int j = 0; j < 8; ++j) acc[t][j] = bv;
  }
  {
    MMSrc s[4] = {{x_cur + (size_t)si * 256, We + 0, 256},
                  {x_cur + (size_t)di * 256, We + 64, 256},
                  {ea_in + (size_t)es * 64, We + 128, 256},
                  {u_cur + (size_t)gi * 256, We + 192, 256}};
    mm16<4, 4>(s, 0, acc, sub, koff);
  }
#pragma unroll
  for (int t = 0; t < 4; ++t)
#pragma unroll
    for (int j = 0; j < 8; ++j)
      lds[wv][j + 8 * half][t * 16 + sub] = fmaxf(acc[t][j], 0.0f);
  asm volatile("s_wait_dscnt 0" ::: "memory");
  const float* eo = &lds[wv][sub][0];
  const float* hrow = ea_in + (size_t)es * 64;

  // ---- GRU: r,z fused as one K=128 GEMM; n kept split ----
  v8f rz[8];
#pragma unroll
  for (int t = 0; t < 8; ++t) {
    float bv = bih[t * 16 + sub] + bhh[t * 16 + sub];
#pragma unroll
    for (int j = 0; j < 8; ++j) rz[t][j] = bv;
  }
  {
    MMSrc s[2] = {{eo, Wih, 64}, {hrow, Whh, 64}};
    mm16<2, 8>(s, 0, rz, sub, koff);
  }
  v8f ni[4], nh[4];
#pragma unroll
  for (int t = 0; t < 4; ++t) {
    float b1 = bih[128 + t * 16 + sub], b2 = bhh[128 + t * 16 + sub];
#pragma unroll
    for (int j = 0; j < 8; ++j) { ni[t][j] = b1; nh[t][j] = b2; }
  }
  {
    MMSrc s1[1] = {{eo, Wih, 64}};
    mm16<1, 4>(s1, 128, ni, sub, koff);
    MMSrc s2[1] = {{hrow, Whh, 64}};
    mm16<1, 4>(s2, 128, nh, sub, koff);
  }

  int dstj[8];
#pragma unroll
  for (int j = 0; j < 8; ++j) dstj[j] = dst[e0 + j + 8 * half];
#pragma unroll
  for (int t = 0; t < 4; ++t) {
    const int c = t * 16 + sub;
#pragma unroll
    for (int j = 0; j < 8; ++j) {
      const size_t e = (size_t)(e0 + j + 8 * half);
      float r = sigmf(rz[t][j]);
      float z = sigmf(rz[t + 4][j]);
      float n = tanhf(ni[t][j] + r * nh[t][j]);
      float h = ea_in[e * 64 + c];
      float hn = (1.0f - z) * n + z * h;
      ea_out[e * 64 + c] = hn;
      atomicAdd(&agg[(size_t)dstj[j] * 64 + c], hn);
    }
  }
}

// ---------------------------------------------------------------------------
// Node step: x_out = relu([x, agg_mean, u[batch]] @ Wn.T + bn); x' = GRU(..)
// ---------------------------------------------------------------------------
__global__ __launch_bounds__(128) void node_step_kernel(
    const float* __restrict__ x_cur, float* __restrict__ x_next,
    const float* __restrict__ u_cur, const float* __restrict__ agg,
    float* __restrict__ gsum, const int* __restrict__ batch,
    const float* __restrict__ Wn, const float* __restrict__ bn,
    const float* __restrict__ Wih, const float* __restrict__ Whh,
    const float* __restrict__ bih, const float* __restrict__ bhh) {
  __shared__ float lds[4][16][68];
  const int lane = threadIdx.x & 31;
  const int wv = threadIdx.x >> 5;
  const int sub = lane & 15;
  const int half = lane >> 4;
  const int koff = half * 2;
  const int tile = blockIdx.x * 4 + wv;
  if (tile * 16 >= NN) return;
  const int i0 = tile * 16;
  const int is = i0 + sub;
  const int gi = batch[is];

  v8f acc[4];
#pragma unroll
  for (int t = 0; t < 4; ++t) {
    float bv = bn[t * 16 + sub];
#pragma unroll
    for (int j = 0; j < 8; ++j) acc[t][j] = bv;
  }
  {
    MMSrc s[3] = {{x_cur + (size_t)is * 256, Wn + 0, 192},
                  {agg + (size_t)is * 64, Wn + 64, 192},
                  {u_cur + (size_t)gi * 256, Wn + 128, 192}};
    mm16<3, 4>(s, 0, acc, sub, koff);
  }
#pragma unroll
  for (int t = 0; t < 4; ++t)
#pragma unroll
    for (int j = 0; j < 8; ++j)
      lds[wv][j + 8 * half][t * 16 + sub] = fmaxf(acc[t][j], 0.0f);
  asm volatile("s_wait_dscnt 0" ::: "memory");
  const float* eo = &lds[wv][sub][0];
  const float* hrow = x_cur + (size_t)is * 256;

  v8f rz[8];
#pragma unroll
  for (int t = 0; t < 8; ++t) {
    float bv = bih[t * 16 + sub] + bhh[t * 16 + sub];
#pragma unroll
    for (int j = 0; j < 8; ++j) rz[t][j] = bv;
  }
  {
    MMSrc s[2] = {{eo, Wih, 64}, {hrow, Whh, 64}};
    mm16<2, 8>(s, 0, rz, sub, koff);
  }
  v8f ni[4], nh[4];
#pragma unroll
  for (int t = 0; t < 4; ++t) {
    float b1 = bih[128 + t * 16 + sub], b2 = bhh[128 + t * 16 + sub];
#pragma unroll
    for (int j = 0; j < 8; ++j) { ni[t][j] = b1; nh[t][j] = b2; }
  }
  {
    MMSrc s1[1] = {{eo, Wih, 64}};
    mm16<1, 4>(s1, 128, ni, sub, koff);
    MMSrc s2[1] = {{hrow, Whh, 64}};
    mm16<1, 4>(s2, 128, nh, sub, koff);
  }

  int bj[8];
#pragma unroll
  for (int j = 0; j < 8; ++j) bj[j] = batch[i0 + j + 8 * half];
#pragma unroll
  for (int t = 0; t < 4; ++t) {
    const int c = t * 16 + sub;
#pragma unroll
    for (int j = 0; j < 8; ++j) {
      const size_t i = (size_t)(i0 + j + 8 * half);
      float r = sigmf(rz[t][j]);
      float z = sigmf(rz[t + 4][j]);
      float n = tanhf(ni[t][j] + r * nh[t][j]);
      float h = x_cur[i * 256 + c];
      float xn = (1.0f - z) * n + z * h;
      x_next[i * 256 + c] = xn;
      atomicAdd(&gsum[(size_t)bj[j] * 64 + c], xn);
    }
  }
}

// ---------------------------------------------------------------------------
// Global step: u_out = relu([mean_x, u] @ Wg.T + bg); u' = GRU(u_out, u)
// 64 graphs -> 4 wave tiles, single block.
// ---------------------------------------------------------------------------
__global__ __launch_bounds__(128) void glob_step_kernel(
    const float* __restrict__ gmean, const float* __restrict__ u_cur,
    float* __restrict__ u_next, const float* __restrict__ Wg,
    const float* __restrict__ bg, const float* __restrict__ Wih,
    const float* __restrict__ Whh, const float* __restrict__ bih,
    const float* __restrict__ bhh) {
  __shared__ float lds[4][16][68];
  const int lane = threadIdx.x & 31;
  const int wv = threadIdx.x >> 5;
  const int sub = lane & 15;
  const int half = lane >> 4;
  const int koff = half * 2;
  const int g0 = wv * 16;
  const int gs = g0 + sub;

  v8f acc[4];
#pragma unroll
  for (int t = 0; t < 4; ++t) {
    float bv = bg[t * 16 + sub];
#pragma unroll
    for (int j = 0; j < 8; ++j) acc[t][j] = bv;
  }
  {
    MMSrc s[2] = {{gmean + (size_t)gs * 64, Wg + 0, 128},
                  {u_cur + (size_t)gs * 256, Wg + 64, 128}};
    mm16<2, 4>(s, 0, acc, sub, koff);
  }
#pragma unroll
  for (int t = 0; t < 4; ++t)
#pragma unroll
    for (int j = 0; j < 8; ++j)
      lds[wv][j + 8 * half][t * 16 + sub] = fmaxf(acc[t][j], 0.0f);
  asm volatile("s_wait_dscnt 0" ::: "memory");
  const float* eo = &lds[wv][sub][0];
  const float* hrow = u_cur + (size_t)gs * 256;

  v8f rz[8];
#pragma unroll
  for (int t = 0; t < 8; ++t) {
    float bv = bih[t * 16 + sub] + bhh[t * 16 + sub];
#pragma unroll
    for (int j = 0; j < 8; ++j) rz[t][j] = bv;
  }
  {
    MMSrc s[2] = {{eo, Wih, 64}, {hrow, Whh, 64}};
    mm16<2, 8>(s, 0, rz, sub, koff);
  }
  v8f ni[4], nh[4];
#pragma unroll
  for (int t = 0; t < 4; ++t) {
    float b1 = bih[128 + t * 16 + sub], b2 = bhh[128 + t * 16 + sub];
#pragma unroll
    for (int j = 0; j < 8; ++j) { ni[t][j] = b1; nh[t][j] = b2; }
  }
  {
    MMSrc s1[1] = {{eo, Wih, 64}};
    mm16<1, 4>(s1, 128, ni, sub, koff);
    MMSrc s2[1] = {{hrow, Whh, 64}};
    mm16<1, 4>(s2, 128, nh, sub, koff);
  }
#pragma unroll
  for (int t = 0; t < 4; ++t) {
    const int c = t * 16 + sub;
#pragma unroll
    for (int j = 0; j < 8; ++j) {
      const size_t g = (size_t)(g0 + j + 8 * half);
      float r = sigmf(rz[t][j]);
      float z = sigmf(rz[t + 4][j]);
      float n = tanhf(ni[t][j] + r * nh[t][j]);
      float h = u_cur[g * 256 + c];
      u_next[g * 256 + c] = (1.0f - z) * n + z * h;
    }
  }
}

// ---------------------------- utility kernels ------------------------------
__global__ void zero_f32(float* p, long n) {
  long i = (long)blockIdx.x * blockDim.x + threadIdx.x;
  long stride = (long)gridDim.x * blockDim.x;
  for (; i < n; i += stride) p[i] = 0.0f;
}
__global__ void copy_to_slab(const float* __restrict__ in, float* __restrict__ out, int rows) {
  int i = blockIdx.x * blockDim.x + threadIdx.x;
  if (i < rows * 64) out[(size_t)(i >> 6) * 256 + (i & 63)] = in[i];
}
__global__ void count_idx(const int* __restrict__ idx, float* cnt, int n) {
  int i = blockIdx.x * blockDim.x + threadIdx.x;
  if (i < n) atomicAdd(&cnt[idx[i]], 1.0f);
}
__global__ void seg_div(float* sums, const float* __restrict__ cnt, int rows) {
  int i = blockIdx.x * blockDim.x + threadIdx.x;
  if (i < rows * 64) sums[i] /= fmaxf(cnt[i >> 6], 1.0f);
}

// ---------------------------------------------------------------------------
extern "C" void kernel_launch(void* const* d_in, const int* in_sizes, int n_in,
                              void* d_out, int out_size, void* d_ws, size_t ws_size,
                              hipStream_t stream) {
  const float* x    = (const float*)d_in[0];
  const int*   eix  = (const int*)d_in[1];
  const float* ea0  = (const float*)d_in[2];
  const float* u    = (const float*)d_in[3];
  const int*   bat  = (const int*)d_in[4];
  const float* We   = (const float*)d_in[6];
  const float* be   = (const float*)d_in[7];
  const float* Wn   = (const float*)d_in[8];
  const float* bn   = (const float*)d_in[9];
  const float* Wg   = (const float*)d_in[10];
  const float* bg   = (const float*)d_in[11];
  const float* eWih = (const float*)d_in[12];
  const float* eWhh = (const float*)d_in[13];
  const float* ebih = (const float*)d_in[14];
  const float* ebhh = (const float*)d_in[15];
  const float* nWih = (const float*)d_in[16];
  const float* nWhh = (const float*)d_in[17];
  const float* nbih = (const float*)d_in[18];
  const float* nbhh = (const float*)d_in[19];
  const float* gWih = (const float*)d_in[20];
  const float* gWhh = (const float*)d_in[21];
  const float* gbih = (const float*)d_in[22];
  const float* gbhh = (const float*)d_in[23];
  const int* src = eix;
  const int* dst = eix + EE;

  float* out_x = (float*)d_out;                      // N x 256 history slab
  float* out_u = (float*)d_out + (size_t)NN * 256;   // G x 256 history slab

  float* ws   = (float*)d_ws;
  float* ea   = ws;  ws += (size_t)EE * 64;   // evolving edge_attr
  float* agg  = ws;  ws += (size_t)NN * 64;   // per-node edge aggregation
  float* ncnt = ws;  ws += NN;                // in-degree counts
  float* gsum = ws;  ws += (size_t)GG * 64;   // per-graph node sums
  float* gcnt = ws;  ws += GG;                // nodes per graph

  // init: histories block 0 and segment counts
  copy_to_slab<<<(NN * 64 + 255) / 256, 256, 0, stream>>>(x, out_x, NN);
  copy_to_slab<<<(GG * 64 + 255) / 256, 256, 0, stream>>>(u, out_u, GG);
  zero_f32<<<256, 256, 0, stream>>>(ncnt, NN);
  zero_f32<<<1, 64, 0, stream>>>(gcnt, GG);
  count_idx<<<(EE + 255) / 256, 256, 0, stream>>>(dst, ncnt, EE);
  count_idx<<<(NN + 255) / 256, 256, 0, stream>>>(bat, gcnt, NN);

  for (int s = 0; s < NSTEPS; ++s) {
    const float* xc = out_x + (size_t)s * 64;
    const float* uc = out_u + (size_t)s * 64;
    zero_f32<<<2048, 256, 0, stream>>>(agg, (long)NN * 64);
    edge_step_kernel<<<EE / 64, 128, 0, stream>>>(
        xc, uc, (s == 0) ? ea0 : ea, ea, agg, src, dst, bat,
        We, be, eWih, eWhh, ebih, ebhh);
    seg_div<<<(NN * 64 + 255) / 256, 256, 0, stream>>>(agg, ncnt, NN);
    zero_f32<<<16, 256, 0, stream>>>(gsum, (long)GG * 64);
    node_step_kernel<<<(NN / 16 + 3) / 4, 128, 0, stream>>>(
        xc, out_x + (size_t)(s + 1) * 64, uc, agg, gsum, bat,
        Wn, bn, nWih, nWhh, nbih, nbhh);
    seg_div<<<(GG * 64 + 255) / 256, 256, 0, stream>>>(gsum, gcnt, GG);
    glob_step_kernel<<<1, 128, 0, stream>>>(
        gsum, uc, out_u + (size_t)(s + 1) * 64,
        Wg, bg, gWih, gWhh, gbih, gbhh);
  }
}